// LogisticATLoss_66881230733470
// MI455X (gfx1250) — compile-verified
//
#include <hip/hip_runtime.h>

#define NUM_Q 9           // NUM_CLASSES - 1
#define REG_HALF 0.005f   // REG_LAMBDA / 2

typedef __attribute__((ext_vector_type(2))) float v2f;
typedef __attribute__((ext_vector_type(8))) float v8f;

// Full 32-lane wave sum. The WMMA (A = ones 16x4, B = [p,0] 4x16) produces
// D[m][n] = p_n + p_{n+16} in every lane's first D VGPR (lane n holds column
// S_{n%16}); four xor-shuffles then sum the 16 distinct column sums.
__device__ __forceinline__ float wave_sum_wmma(float p) {
  v2f a; a.x = 1.0f; a.y = 1.0f;          // A = ones(16x4)
  v2f b; b.x = p;    b.y = 0.0f;          // B rows 0/2 = lane partials, rows 1/3 = 0
  v8f c = {};
  c = __builtin_amdgcn_wmma_f32_16x16x4_f32(
      /*neg_a=*/false, a, /*neg_b=*/false, b,
      /*c_mod=*/(short)0, c, /*reuse_a=*/false, /*reuse_b=*/false);
  float s = c[0];                          // S_{lane & 15}
  s += __shfl_xor(s, 1, 32);
  s += __shfl_xor(s, 2, 32);
  s += __shfl_xor(s, 4, 32);
  s += __shfl_xor(s, 8, 32);
  return s;                                // sum over all 32 lanes
}

// Per-element loss: Sum_q softplus(s_q * (thr_q - p)), s_q = (q < t) ? +1 : -1
//  = 0.5*(Sum|d_q| + (A[t] + p*(9-2t))) + log( Prod_q (1 + min(e^{thr_q}e^{-p}, e^{-thr_q}e^{p})) )
__device__ __forceinline__ float elem_loss(float w, float xi, int ti,
                                           const float* __restrict__ thrv,
                                           const float* __restrict__ eP,
                                           const float* __restrict__ eM,
                                           const float* s_A) {
  float p  = w * xi;
  float em = __expf(-p);
  float ep = __expf(p);
  float absacc = 0.0f;
  float prod   = 1.0f;
#pragma unroll
  for (int q = 0; q < NUM_Q; ++q) {
    float d = thrv[q] - p;
    absacc += fabsf(d);                        // |d| is a free VALU src modifier
    float u = fminf(eP[q] * em, eM[q] * ep);   // = exp(-|d_q|), in (0,1]
    prod = __builtin_fmaf(prod, u, prod);      // prod *= (1 + u)
  }
  float sig = __builtin_fmaf(p, (float)(NUM_Q - 2 * ti), s_A[ti]); // Sum s_q d_q
  return 0.5f * (absacc + sig) + __logf(prod);
}

__global__ void logistic_at_loss_kernel(const float* __restrict__ w1,
                                        const float* __restrict__ x,
                                        const float* __restrict__ thr,
                                        const int*   __restrict__ tgt,
                                        double*      __restrict__ wsum,
                                        int N) {
  __shared__ float s_eP[NUM_Q];
  __shared__ float s_eM[NUM_Q];
  __shared__ float s_A[NUM_Q + 1];
  __shared__ float s_wave[8];

  const int lane = threadIdx.x & 31;
  const int wid  = threadIdx.x >> 5;

  // Per-block constant tables: e^{+-thr_q} and the signed-prefix table
  // A[t] = 2*T_t - S  (T_t = sum_{q<t} thr_q, S = sum_q thr_q).
  if (threadIdx.x < NUM_Q) {
    float t = thr[threadIdx.x];
    s_eP[threadIdx.x] = __expf(t);
    s_eM[threadIdx.x] = __expf(-t);
  }
  if (threadIdx.x == 0) {
    float S = 0.0f;
    for (int q = 0; q < NUM_Q; ++q) S += thr[q];
    float T = 0.0f;
    for (int t = 0; t <= NUM_Q; ++t) {
      s_A[t] = 2.0f * T - S;
      if (t < NUM_Q) T += thr[t];
    }
  }
  __syncthreads();

  float thrv[NUM_Q], eP[NUM_Q], eM[NUM_Q];
#pragma unroll
  for (int q = 0; q < NUM_Q; ++q) {
    thrv[q] = thr[q];      // uniform -> scalar loads
    eP[q]   = s_eP[q];
    eM[q]   = s_eM[q];
  }

  const float w = w1[0];
  float acc = 0.0f;

  const int nvec = N >> 2;
  const float4* __restrict__ x4 = (const float4*)x;
  const int4*   __restrict__ t4 = (const int4*)tgt;
  const int stride = gridDim.x * blockDim.x;
  const int gid    = blockIdx.x * blockDim.x + threadIdx.x;

  for (int i = gid; i < nvec; i += stride) {
    __builtin_prefetch(&x4[i + stride], 0, 0);   // global_prefetch_b8 (speculative)
    __builtin_prefetch(&t4[i + stride], 0, 0);
    float4 xv = x4[i];
    int4   tv = t4[i];
    acc += elem_loss(w, xv.x, tv.x, thrv, eP, eM, s_A);
    acc += elem_loss(w, xv.y, tv.y, thrv, eP, eM, s_A);
    acc += elem_loss(w, xv.z, tv.z, thrv, eP, eM, s_A);
    acc += elem_loss(w, xv.w, tv.w, thrv, eP, eM, s_A);
  }

  // Scalar tail (N % 4 elements) — reconverges before the WMMA reduction.
  const int base = nvec << 2;
  const int rem  = N - base;
  if (gid < rem) acc += elem_loss(w, x[base + gid], tgt[base + gid], thrv, eP, eM, s_A);

  // Wave reduction via V_WMMA_F32_16X16X4_F32 (+4 xor shuffles), then block
  // reduction in LDS, then one f64 global atomic per block.
  float ws = wave_sum_wmma(acc);
  if (lane == 0) s_wave[wid] = ws;
  __syncthreads();
  if (threadIdx.x == 0) {
    float b = 0.0f;
    const int nw = blockDim.x >> 5;
    for (int i = 0; i < nw; ++i) b += s_wave[i];
    atomicAdd(wsum, (double)b);                  // global_atomic_add_f64
  }
}

__global__ void finalize_kernel(const double* __restrict__ wsum,
                                const float*  __restrict__ w1,
                                float* __restrict__ out) {
  float w = w1[0];
  out[0] = (float)wsum[0] + REG_HALF * w * w;
}

extern "C" void kernel_launch(void* const* d_in, const int* in_sizes, int n_in,
                              void* d_out, int out_size, void* d_ws, size_t ws_size,
                              hipStream_t stream) {
  const float* wts = (const float*)d_in[0];   // weights [1]
  const float* x   = (const float*)d_in[1];   // x [N]
  const float* thr = (const float*)d_in[2];   // thresholds [9]
  const int*   tgt = (const int*)d_in[3];     // target [N]
  const int N = in_sizes[1];

  double* wsum = (double*)d_ws;
  hipMemsetAsync(wsum, 0, sizeof(double), stream);   // graph-capturable

  const int threads = 256;
  int nvec = N >> 2;
  int blocks = (nvec + threads - 1) / threads;
  if (blocks > 2048) blocks = 2048;   // grid-stride; 16K waves saturates HBM
  if (blocks < 1) blocks = 1;

  logistic_at_loss_kernel<<<blocks, threads, 0, stream>>>(wts, x, thr, tgt, wsum, N);
  finalize_kernel<<<1, 1, 0, stream>>>(wsum, wts, (float*)d_out);
}